// GraphNeuralNetworkLayer_42150809042945
// MI455X (gfx1250) — compile-verified
//
#include <hip/hip_runtime.h>
#include <hip/hip_bf16.h>

#define D 64
#define EPS_BN 1e-5f

typedef __attribute__((ext_vector_type(2)))  float v2f;
typedef __attribute__((ext_vector_type(8)))  float v8f;

// ---------------------------------------------------------------------------
// Degree accumulation: deg[col[e]] += 1
// ---------------------------------------------------------------------------
__global__ __launch_bounds__(256) void k_deg(const int* __restrict__ col,
                                             float* __restrict__ deg, int E) {
  int e = blockIdx.x * blockDim.x + threadIdx.x;
  if (e < E) atomicAdd(&deg[col[e]], 1.0f);
}

// dinv[i] = rsqrt(deg[i] + 1)   (in-place over deg buffer)
__global__ __launch_bounds__(256) void k_dinv(float* __restrict__ deg, int n) {
  int i = blockIdx.x * blockDim.x + threadIdx.x;
  if (i < n) deg[i] = rsqrtf(deg[i] + 1.0f);
}

// ---------------------------------------------------------------------------
// H = X(nrows x 64) @ W(64 x 64) using V_WMMA_F32_16X16X4_F32.
// One wave -> 16 rows x 64 cols (four 16x16 accumulators), K-loop step 4.
// 256 threads = 8 waves = 128 rows per block. nrows % 16 == 0 expected
// (100000 = 6250*16) so EXEC stays all-ones for every WMMA.
// ---------------------------------------------------------------------------
__global__ __launch_bounds__(256) void k_gemm_wmma(const float* __restrict__ X,
                                                   const float* __restrict__ W,
                                                   float* __restrict__ H,
                                                   int nrows) {
  const int wave = threadIdx.x >> 5;
  const int lane = threadIdx.x & 31;
  const int half = lane >> 4;   // 0: lanes 0-15, 1: lanes 16-31
  const int lid  = lane & 15;
  const int r0   = (blockIdx.x * 8 + wave) * 16;
  if (r0 >= nrows) return;      // uniform per-wave exit

  v8f acc0 = {}, acc1 = {}, acc2 = {}, acc3 = {};

  // A-matrix 16x4 f32 layout: lane(l<16) row=r0+l, VGPR0/1 = K {0,1};
  // lanes 16-31 same rows, K {2,3}  ->  contiguous v2f at column k + 2*half.
  const float* xrow = X + (size_t)(r0 + lid) * D;

  for (int k = 0; k < D; k += 4) {
    v2f a = *(const v2f*)(xrow + k + 2 * half);

    // B-matrix 4x16 f32: K rows striped like A across lane halves,
    // N = n0 + lid across lanes 0-15 of each half.
    const float* wk0 = W + (size_t)(k + 2 * half) * D + lid; // K row (even)
    const float* wk1 = wk0 + D;                              // K row (odd)
    v2f b0, b1, b2, b3;
    b0.x = wk0[0];  b0.y = wk1[0];
    b1.x = wk0[16]; b1.y = wk1[16];
    b2.x = wk0[32]; b2.y = wk1[32];
    b3.x = wk0[48]; b3.y = wk1[48];

#if defined(__HIP_DEVICE_COMPILE__) && __has_builtin(__builtin_amdgcn_wmma_f32_16x16x4_f32)
    acc0 = __builtin_amdgcn_wmma_f32_16x16x4_f32(false, a, false, b0, (short)0, acc0, false, false);
    acc1 = __builtin_amdgcn_wmma_f32_16x16x4_f32(false, a, false, b1, (short)0, acc1, false, false);
    acc2 = __builtin_amdgcn_wmma_f32_16x16x4_f32(false, a, false, b2, (short)0, acc2, false, false);
    acc3 = __builtin_amdgcn_wmma_f32_16x16x4_f32(false, a, false, b3, (short)0, acc3, false, false);
#else
    // parse-only fallback (host pass / builtin unavailable)
    acc0 += a.x * b0.x; acc1 += a.x * b1.x; acc2 += a.x * b2.x; acc3 += a.x * b3.x;
#endif
  }

  // C/D layout: VGPR j -> row r0 + j + 8*half, col n0 + lid
  float* hrow = H + (size_t)(r0 + 8 * half) * D + lid;
  #pragma unroll
  for (int j = 0; j < 8; ++j) {
    hrow[(size_t)j * D + 0]  = acc0[j];
    hrow[(size_t)j * D + 16] = acc1[j];
    hrow[(size_t)j * D + 32] = acc2[j];
    hrow[(size_t)j * D + 48] = acc3[j];
  }
}

// ---------------------------------------------------------------------------
// Edge scatter: agg[col[e], :] += H[row[e], :] * dinv[row]*dinv[col]
// 16 threads per edge, float4 per thread -> 4 global f32 atomics each.
// ---------------------------------------------------------------------------
__global__ __launch_bounds__(256) void k_scatter(const float* __restrict__ H,
                                                 const int* __restrict__ row,
                                                 const int* __restrict__ col,
                                                 const float* __restrict__ dinv,
                                                 float* __restrict__ agg, int E) {
  long long tid = (long long)blockIdx.x * blockDim.x + threadIdx.x;
  int e = (int)(tid >> 4);
  if (e >= E) return;
  int f = (int)(tid & 15) << 2;
  int r = row[e], c = col[e];
  float w = dinv[r] * dinv[c];
  const float4 v = *(const float4*)(H + (size_t)r * D + f);
  float* dst = agg + (size_t)c * D + f;
  atomicAdd(dst + 0, v.x * w);
  atomicAdd(dst + 1, v.y * w);
  atomicAdd(dst + 2, v.z * w);
  atomicAdd(dst + 3, v.w * w);
}

// agg += H * self_norm + bias
__global__ __launch_bounds__(256) void k_self_bias(float* __restrict__ agg,
                                                   const float* __restrict__ H,
                                                   const float* __restrict__ dinv,
                                                   const float* __restrict__ b, int n) {
  int idx = blockIdx.x * blockDim.x + threadIdx.x;
  if (idx >= n * D) return;
  int i = idx >> 6, f = idx & 63;
  float di = dinv[i];
  agg[idx] += H[idx] * di * di + b[f];
}

// ---------------------------------------------------------------------------
// BN statistics: stats[0:64] = sum_f, stats[64:128] = sumsq_f
// ---------------------------------------------------------------------------
__global__ __launch_bounds__(256) void k_bn_stats(const float* __restrict__ H,
                                                  float* __restrict__ stats, int n) {
  __shared__ float ssum[256], ssq[256];
  int t = threadIdx.x;
  int f = t & 63;
  int p = t >> 6;  // 0..3 rows in flight per block
  float sum = 0.f, sq = 0.f;
  for (int r = blockIdx.x * 4 + p; r < n; r += gridDim.x * 4) {
    float v = H[(size_t)r * D + f];
    sum += v; sq += v * v;
  }
  ssum[t] = sum; ssq[t] = sq;
  __syncthreads();
  if (t < 128) { ssum[t] += ssum[t + 128]; ssq[t] += ssq[t + 128]; }
  __syncthreads();
  if (t < 64) {
    atomicAdd(&stats[t],      ssum[t] + ssum[t + 64]);
    atomicAdd(&stats[64 + t], ssq[t]  + ssq[t + 64]);
  }
}

__global__ __launch_bounds__(256) void k_bn_apply(const float* __restrict__ Hin,
                                                  const float* __restrict__ stats,
                                                  const float* __restrict__ g,
                                                  const float* __restrict__ bt,
                                                  float* __restrict__ out,
                                                  int n, float inv_n, int relu) {
  int idx = blockIdx.x * blockDim.x + threadIdx.x;
  if (idx >= n * D) return;
  int f = idx & 63;
  float mean = stats[f] * inv_n;
  float var  = stats[64 + f] * inv_n - mean * mean;   // biased var
  float v = (Hin[idx] - mean) * rsqrtf(var + EPS_BN) * g[f] + bt[f];
  if (relu) v = fmaxf(v, 0.f);
  out[idx] = v;
}

// ---------------------------------------------------------------------------
extern "C" void kernel_launch(void* const* d_in, const int* in_sizes, int n_in,
                              void* d_out, int out_size, void* d_ws, size_t ws_size,
                              hipStream_t stream) {
  const float* x   = (const float*)d_in[0];
  const int*   ei  = (const int*)d_in[1];
  const float* W0  = (const float*)d_in[2];
  const float* b0  = (const float*)d_in[3];
  const float* g0  = (const float*)d_in[4];
  const float* bt0 = (const float*)d_in[5];
  const float* W1  = (const float*)d_in[6];
  const float* b1  = (const float*)d_in[7];
  const float* g1  = (const float*)d_in[8];
  const float* bt1 = (const float*)d_in[9];

  const int n = in_sizes[0] / D;        // 100000
  const int E = in_sizes[1] / 2;        // 1200000
  const int* row = ei;
  const int* col = ei + E;
  float* out = (float*)d_out;

  // workspace: dinv[n] | A[n*D] | B[n*D] | stats[128]   (~52 MB)
  float* dinv  = (float*)d_ws;
  float* A     = dinv + n;
  float* B     = A + (size_t)n * D;
  float* stats = B + (size_t)n * D;

  const int nd      = n * D;
  const int gElem   = (nd + 255) / 256;
  const int gEdge   = (E + 255) / 256;
  const int gScat   = (int)(((long long)E * 16 + 255) / 256);
  const int gGemm   = ((n + 15) / 16 + 7) / 8;
  const float inv_n = 1.0f / (float)n;

  // ---- norms ----
  hipMemsetAsync(dinv, 0, (size_t)n * sizeof(float), stream);
  k_deg <<<gEdge, 256, 0, stream>>>(col, dinv, E);
  k_dinv<<<(n + 255) / 256, 256, 0, stream>>>(dinv, n);

  // ---- layer 1 ----
  k_gemm_wmma<<<gGemm, 256, 0, stream>>>(x, W0, A, n);
  hipMemsetAsync(B, 0, (size_t)nd * sizeof(float), stream);
  k_scatter  <<<gScat, 256, 0, stream>>>(A, row, col, dinv, B, E);
  k_self_bias<<<gElem, 256, 0, stream>>>(B, A, dinv, b0, n);
  hipMemsetAsync(stats, 0, 128 * sizeof(float), stream);
  k_bn_stats <<<256, 256, 0, stream>>>(B, stats, n);
  k_bn_apply <<<gElem, 256, 0, stream>>>(B, stats, g0, bt0, A, n, inv_n, 1);

  // ---- layer 2 ----
  k_gemm_wmma<<<gGemm, 256, 0, stream>>>(A, W1, B, n);
  hipMemsetAsync(out, 0, (size_t)nd * sizeof(float), stream);
  k_scatter  <<<gScat, 256, 0, stream>>>(B, row, col, dinv, out, E);
  k_self_bias<<<gElem, 256, 0, stream>>>(out, B, dinv, b1, n);
  hipMemsetAsync(stats, 0, 128 * sizeof(float), stream);
  k_bn_stats <<<256, 256, 0, stream>>>(out, stats, n);
  k_bn_apply <<<gElem, 256, 0, stream>>>(out, stats, g1, bt1, out, n, inv_n, 0);
}